// LINEAR_68436008894573
// MI455X (gfx1250) — compile-verified
//
#include <hip/hip_runtime.h>

// Linear RNN h_t = h_{t-1} @ W_hh^T + proj_{t-1}, chunked parallel scan.
// N=32 T=2048 I=256 H=512. All math fp32 via V_WMMA_F32_16X16X4_F32 (CDNA5).
// Workspace requirement: (H*H + 64*H*H + 32*N*H) * 4 bytes ~= 71 MB.

#define NB 32
#define TT 2048
#define II 256
#define HH 512
#define CHUNK 64
#define JCH (TT / CHUNK)          // 32 chunks
#define PITCH 516                 // LDS row pitch (floats), conflict-spread
#define SMEM_BYTES (NB * PITCH * 4)

typedef __attribute__((ext_vector_type(2))) float v2f;
typedef __attribute__((ext_vector_type(8))) float v8f;

__device__ __forceinline__ v8f wmma_f32(v2f a, v2f b, v8f c) {
  // D = A(16x4) * B(4x16) + C(16x16), fp32 WMMA
  return __builtin_amdgcn_wmma_f32_16x16x4_f32(false, a, false, b, (short)0, c,
                                               false, false);
}

__device__ __forceinline__ v2f ld2(const float* p) {
  return *reinterpret_cast<const v2f*>(p);
}

// One wave computes a 16(M) x 64(N) fp32 tile. All pointers are pre-offset
// per lane: aptr = &A[row][kb], bj = &Wgt[col_j][kb] where the GEMM computed
// is C[m,n] = sum_k A[m,k] * Wgt[n,k]  (i.e. A @ Wgt^T).
// A 16x4 layout: lanes 0-15 rows M, VGPR0/1 = K+0/K+1 (lanes 16-31: K+2/K+3).
__device__ __forceinline__ void wave_gemm(const float* aptr, const float* b0,
                                          const float* b1, const float* b2,
                                          const float* b3, int K, v8f acc[4]) {
#pragma unroll 4
  for (int k0 = 0; k0 < K; k0 += 4) {
    v2f a = ld2(aptr + k0);
    acc[0] = wmma_f32(a, ld2(b0 + k0), acc[0]);
    acc[1] = wmma_f32(a, ld2(b1 + k0), acc[1]);
    acc[2] = wmma_f32(a, ld2(b2 + k0), acc[2]);
    acc[3] = wmma_f32(a, ld2(b3 + k0), acc[3]);
  }
}

// ---------------------------------------------------------------- transpose
__global__ void __launch_bounds__(256) rnn_transpose_kernel(
    const float* __restrict__ W, float* __restrict__ Wt) {
  int idx = blockIdx.x * 256 + threadIdx.x;  // < H*H
  int m = idx >> 9;
  int c = idx & (HH - 1);
  Wt[c * HH + m] = W[idx];
}

// ------------------------------------------------- proj = x @ W_ih^T + b_ih
// proj stored TIME-MAJOR [t][n][h] (row r = t*NB + n) into out1 scratch.
__global__ void __launch_bounds__(128) rnn_proj_kernel(
    const float* __restrict__ x, const float* __restrict__ W_ih,
    const float* __restrict__ b_ih, float* __restrict__ proj) {
  const int lane = threadIdx.x & 31;
  const int wid = threadIdx.x >> 5;          // 4 waves / block
  const int lrow = lane & 15;
  const int kb = (lane >> 4) << 1;           // 0 or 2
  const int rowadd = (lane >> 4) << 3;       // 0 or 8 (C-tile row offset)

  const int mbase = blockIdx.x << 4;                 // 4096 M-tiles
  const int nbase = (blockIdx.y << 8) + (wid << 6);  // 2*4 supertiles of 64

  const int r = mbase + lrow;                // time-major row
  const int tt = r >> 5;                     // r = t*NB + n
  const int nn = r & (NB - 1);
  const float* aptr = x + ((size_t)nn * TT + tt) * II + kb;
  const float* b0 = W_ih + (size_t)(nbase + 0 + lrow) * II + kb;
  const float* b1 = W_ih + (size_t)(nbase + 16 + lrow) * II + kb;
  const float* b2 = W_ih + (size_t)(nbase + 32 + lrow) * II + kb;
  const float* b3 = W_ih + (size_t)(nbase + 48 + lrow) * II + kb;

  v8f acc[4];
#pragma unroll
  for (int jj = 0; jj < 4; ++jj) {
    float bias = b_ih[nbase + jj * 16 + lrow];
#pragma unroll
    for (int v = 0; v < 8; ++v) acc[jj][v] = bias;
  }
  wave_gemm(aptr, b0, b1, b2, b3, II, acc);

#pragma unroll
  for (int jj = 0; jj < 4; ++jj)
#pragma unroll
    for (int v = 0; v < 8; ++v)
      proj[(size_t)(mbase + rowadd + v) * HH + nbase + jj * 16 + lrow] =
          acc[jj][v];
}

// ------------------------------------------------ powers: Out = A @ Wt^T
// With Wt = W_hh^T this computes Out = A @ W_hh  (matrix-power chain).
__global__ void __launch_bounds__(128) rnn_gemm512_kernel(
    const float* __restrict__ A, const float* __restrict__ Wt,
    float* __restrict__ Out) {
  const int lane = threadIdx.x & 31;
  const int wid = threadIdx.x >> 5;
  const int lrow = lane & 15;
  const int kb = (lane >> 4) << 1;
  const int rowadd = (lane >> 4) << 3;
  const int mbase = blockIdx.x << 4;                 // 32 M-tiles
  const int nbase = (blockIdx.y << 8) + (wid << 6);  // 2*4 supertiles

  const float* aptr = A + (size_t)(mbase + lrow) * HH + kb;
  const float* b0 = Wt + (size_t)(nbase + 0 + lrow) * HH + kb;
  const float* b1 = Wt + (size_t)(nbase + 16 + lrow) * HH + kb;
  const float* b2 = Wt + (size_t)(nbase + 32 + lrow) * HH + kb;
  const float* b3 = Wt + (size_t)(nbase + 48 + lrow) * HH + kb;

  v8f acc[4] = {};
  wave_gemm(aptr, b0, b1, b2, b3, HH, acc);
#pragma unroll
  for (int jj = 0; jj < 4; ++jj)
#pragma unroll
    for (int v = 0; v < 8; ++v)
      Out[(size_t)(mbase + rowadd + v) * HH + nbase + jj * 16 + lrow] =
          acc[jj][v];
}

// ----------------------- phase A: per-chunk local recurrence, u kept in LDS
// One workgroup (16 waves) per chunk j. u_{jC} = proj[jC-1] (j>0) or
// initial + proj[0] (j==0); then C-1 steps u_t = u_{t-1} @ W_hh^T + proj[t-1].
// Writes u to out0 in [n][t][h] layout. The next step's proj slice is
// prefetched (global_prefetch_b8) while the current GEMM runs so the
// post-barrier accumulator init doesn't stall the serial critical path.
__global__ void __launch_bounds__(512) rnn_scanA_kernel(
    const float* __restrict__ proj, const float* __restrict__ initial,
    const float* __restrict__ W_hh, float* __restrict__ u) {
  extern __shared__ float sh[];  // NB x PITCH floats (~66 KB of 320 KB LDS)
  const int j = blockIdx.x;
  const int t0 = j * CHUNK;
  const int tid = threadIdx.x;

  {  // first element of the chunk (no GEMM)
    const float* src = (j == 0) ? proj : (proj + (size_t)(t0 - 1) * NB * HH);
    for (int idx = tid; idx < NB * HH; idx += 512) {
      int n = idx >> 9, h = idx & (HH - 1);
      float v = src[idx];
      if (j == 0) v += initial[idx];
      sh[n * PITCH + h] = v;
      u[((size_t)n * TT + t0) * HH + h] = v;
    }
  }
  __syncthreads();

  const int lane = tid & 31;
  const int wid = tid >> 5;                  // 16 waves: 2 M-tiles x 8 N-supers
  const int lrow = lane & 15;
  const int kb = (lane >> 4) << 1;
  const int rowadd = (lane >> 4) << 3;
  const int mbase = (wid & 1) << 4;
  const int nbase = (wid >> 1) << 6;

  const float* aptr = sh + (mbase + lrow) * PITCH + kb;  // LDS A operand
  const float* b0 = W_hh + (size_t)(nbase + 0 + lrow) * HH + kb;
  const float* b1 = W_hh + (size_t)(nbase + 16 + lrow) * HH + kb;
  const float* b2 = W_hh + (size_t)(nbase + 32 + lrow) * HH + kb;
  const float* b3 = W_hh + (size_t)(nbase + 48 + lrow) * HH + kb;

  for (int s = 1; s < CHUNK; ++s) {
    const int t = t0 + s;
    const float* q = proj + (size_t)(t - 1) * NB * HH;
    v8f acc[4];
#pragma unroll
    for (int jj = 0; jj < 4; ++jj)
#pragma unroll
      for (int v = 0; v < 8; ++v)
        acc[jj][v] = q[(mbase + rowadd + v) * HH + nbase + jj * 16 + lrow];

    // Prefetch next step's q slice (64 KB = 512 threads * 128 B) into cache
    // while this step's 512 WMMAs execute. Last iteration prefetches the
    // final in-bounds slice (t <= T-1), so addresses are always valid.
    __builtin_prefetch(proj + (size_t)t * NB * HH + tid * 32, 0, 1);

    wave_gemm(aptr, b0, b1, b2, b3, HH, acc);

    __syncthreads();  // all waves finished reading u_{t-1} from LDS
#pragma unroll
    for (int jj = 0; jj < 4; ++jj)
#pragma unroll
      for (int v = 0; v < 8; ++v) {
        int m = mbase + rowadd + v;
        int c = nbase + jj * 16 + lrow;
        float val = acc[jj][v];
        sh[m * PITCH + c] = val;
        u[((size_t)m * TT + t) * HH + c] = val;
      }
    __syncthreads();
  }
}

// ------------- boundary states: b_1 = u_{C-1}; b_j = u_{jC-1} + W^C b_{j-1}
__global__ void __launch_bounds__(512) rnn_boundary_kernel(
    const float* __restrict__ u, const float* __restrict__ PC,  // = W^C
    float* __restrict__ bbuf) {
  extern __shared__ float sh[];
  const int tid = threadIdx.x;
  for (int idx = tid; idx < NB * HH; idx += 512) {
    int n = idx >> 9, h = idx & (HH - 1);
    float v = u[((size_t)n * TT + (CHUNK - 1)) * HH + h];
    sh[n * PITCH + h] = v;
    bbuf[(size_t)NB * HH + idx] = v;  // slot j=1
  }
  __syncthreads();

  const int lane = tid & 31;
  const int wid = tid >> 5;
  const int lrow = lane & 15;
  const int kb = (lane >> 4) << 1;
  const int rowadd = (lane >> 4) << 3;
  const int mbase = (wid & 1) << 4;
  const int nbase = (wid >> 1) << 6;

  const float* aptr = sh + (mbase + lrow) * PITCH + kb;
  const float* b0 = PC + (size_t)(nbase + 0 + lrow) * HH + kb;
  const float* b1 = PC + (size_t)(nbase + 16 + lrow) * HH + kb;
  const float* b2 = PC + (size_t)(nbase + 32 + lrow) * HH + kb;
  const float* b3 = PC + (size_t)(nbase + 48 + lrow) * HH + kb;

  for (int j = 2; j < JCH; ++j) {
    const int tq = j * CHUNK - 1;
    v8f acc[4];
#pragma unroll
    for (int jj = 0; jj < 4; ++jj)
#pragma unroll
      for (int v = 0; v < 8; ++v)
        acc[jj][v] = u[((size_t)(mbase + rowadd + v) * TT + tq) * HH + nbase +
                       jj * 16 + lrow];

    wave_gemm(aptr, b0, b1, b2, b3, HH, acc);

    __syncthreads();
#pragma unroll
    for (int jj = 0; jj < 4; ++jj)
#pragma unroll
      for (int v = 0; v < 8; ++v) {
        int m = mbase + rowadd + v;
        int c = nbase + jj * 16 + lrow;
        float val = acc[jj][v];
        sh[m * PITCH + c] = val;
        bbuf[(size_t)j * NB * HH + (size_t)m * HH + c] = val;
      }
    __syncthreads();
  }
}

// ------------- correction: h_t = u_t + b_j @ (W^{i+1})^T ; duplicate output
// Final stores are non-temporal (write-once, never re-read) so the 256 MB
// output stream doesn't thrash the L2-resident power matrices.
__global__ void __launch_bounds__(512) rnn_correct_kernel(
    const float* __restrict__ bbuf, const float* __restrict__ powers,
    float* __restrict__ out0, float* __restrict__ out1) {
  const int t = blockIdx.x;
  const int j = t >> 6;             // CHUNK = 64
  const int i = t & (CHUNK - 1);
  const int tid = threadIdx.x;

  if (j == 0) {  // chunk 0 is exact: just duplicate
    for (int idx = tid; idx < NB * HH; idx += 512) {
      int n = idx >> 9, h = idx & (HH - 1);
      size_t o = ((size_t)n * TT + t) * HH + h;
      __builtin_nontemporal_store(out0[o], &out1[o]);
    }
    return;
  }

  const int lane = tid & 31;
  const int wid = tid >> 5;
  const int lrow = lane & 15;
  const int kb = (lane >> 4) << 1;
  const int rowadd = (lane >> 4) << 3;
  const int mbase = (wid & 1) << 4;
  const int nbase = (wid >> 1) << 6;

  const float* A = bbuf + (size_t)j * NB * HH;          // b_j  [32 x 512]
  const float* P = powers + (size_t)i * HH * HH;        // W^{i+1}
  const float* aptr = A + (size_t)(mbase + lrow) * HH + kb;
  const float* b0 = P + (size_t)(nbase + 0 + lrow) * HH + kb;
  const float* b1 = P + (size_t)(nbase + 16 + lrow) * HH + kb;
  const float* b2 = P + (size_t)(nbase + 32 + lrow) * HH + kb;
  const float* b3 = P + (size_t)(nbase + 48 + lrow) * HH + kb;

  v8f acc[4];
#pragma unroll
  for (int jj = 0; jj < 4; ++jj)
#pragma unroll
    for (int v = 0; v < 8; ++v)
      acc[jj][v] = out0[((size_t)(mbase + rowadd + v) * TT + t) * HH + nbase +
                        jj * 16 + lrow];

  wave_gemm(aptr, b0, b1, b2, b3, HH, acc);

#pragma unroll
  for (int jj = 0; jj < 4; ++jj)
#pragma unroll
    for (int v = 0; v < 8; ++v) {
      size_t o = ((size_t)(mbase + rowadd + v) * TT + t) * HH + nbase +
                 jj * 16 + lrow;
      float val = acc[jj][v];
      __builtin_nontemporal_store(val, &out0[o]);
      __builtin_nontemporal_store(val, &out1[o]);
    }
}

extern "C" void kernel_launch(void* const* d_in, const int* in_sizes, int n_in,
                              void* d_out, int out_size, void* d_ws,
                              size_t ws_size, hipStream_t stream) {
  const float* x = (const float*)d_in[0];        // [N,T,I]
  const float* initial = (const float*)d_in[1];  // [N,H]
  const float* W_ih = (const float*)d_in[2];     // [H,I]
  const float* b_ih = (const float*)d_in[3];     // [H]
  const float* W_hh = (const float*)d_in[4];     // [H,H]

  float* out0 = (float*)d_out;                   // hiddens copy 0 [N,T,H]
  float* out1 = out0 + (size_t)NB * TT * HH;     // copy 1; doubles as proj
  float* proj = out1;  // staged time-major [T][N][H], overwritten at the end

  float* ws = (float*)d_ws;
  float* Wt = ws;                                   // H*H     (W_hh^T)
  float* powers = Wt + (size_t)HH * HH;             // CHUNK*H*H  (W^1..W^C)
  float* bbuf = powers + (size_t)CHUNK * HH * HH;   // JCH*N*H  boundary states

  // 1) W_hh^T (for the power chain)
  rnn_transpose_kernel<<<(HH * HH) / 256, 256, 0, stream>>>(W_hh, Wt);

  // 2) proj = x @ W_ih^T + b (time-major, into out1 scratch)
  rnn_proj_kernel<<<dim3((TT * NB) / 16, HH / 256), 128, 0, stream>>>(
      x, W_ih, b_ih, proj);

  // 3) powers[k] = W^{k+1}:  P_0 = W_hh;  P_k = P_{k-1} @ W_hh
  hipMemcpyAsync(powers, W_hh, (size_t)HH * HH * sizeof(float),
                 hipMemcpyDeviceToDevice, stream);
  for (int k = 1; k < CHUNK; ++k)
    rnn_gemm512_kernel<<<dim3(HH / 16, HH / 256), 128, 0, stream>>>(
        powers + (size_t)(k - 1) * HH * HH, Wt, powers + (size_t)k * HH * HH);

  // 4) chunk-local recurrences (parallel over 32 chunks, LDS-resident state)
  rnn_scanA_kernel<<<JCH, 512, SMEM_BYTES, stream>>>(proj, initial, W_hh,
                                                     out0);

  // 5) boundary recurrence across chunks (single workgroup, 30 steps)
  rnn_boundary_kernel<<<1, 512, SMEM_BYTES, stream>>>(
      out0, powers + (size_t)(CHUNK - 1) * HH * HH, bbuf);

  // 6) correction + duplicate output (fully parallel over T)
  rnn_correct_kernel<<<TT, 512, 0, stream>>>(bbuf, powers, out0, out1);
}